// MsDroidNet_59889023975523
// MI455X (gfx1250) — compile-verified
//
#include <hip/hip_runtime.h>
#include <math.h>

#define DD 64
#define TSTR 68            // padded LDS row stride (floats) to dodge bank conflicts
#define BN_EPS 1e-5f

typedef __attribute__((ext_vector_type(2))) float v2f;
typedef __attribute__((ext_vector_type(8))) float v8f;

static __device__ inline v8f wmma_f32_16x16x4(v2f a, v2f b, v8f c) {
  // fp32 WMMA: D = A(16x4) x B(4x16) + C(16x16)
  return __builtin_amdgcn_wmma_f32_16x16x4_f32(
      /*neg_a=*/false, a, /*neg_b=*/false, b,
      /*c_mod=*/(short)0, c, /*reuse_a=*/false, /*reuse_b=*/false);
}

// ---------------------------------------------------------------------------
// Edge aggregation: agg[dst][c..c+3] += x[src][c..c+3]
// One thread per (edge, float4 chunk): b128 gather + 4 scatter atomics.
// Working set (x: 25.6 MB, agg: 25.6 MB, edges: 10 MB) is L2-resident;
// the limit is atomic issue rate, so minimize non-atomic overhead per atomic.
// ---------------------------------------------------------------------------
__global__ __launch_bounds__(256) void agg_kernel(
    const float4* __restrict__ x4, const int* __restrict__ ei,
    float* __restrict__ agg, int n_edges) {
  int idx = blockIdx.x * 256 + threadIdx.x;     // total = E * 16
  if (idx >= n_edges * 16) return;
  int e = idx >> 4;
  int c = idx & 15;                             // float4 chunk within the row
  int src = ei[e];
  int dst = ei[n_edges + e];
  float4 v = x4[src * 16 + c];
  float* d = &agg[dst * DD + c * 4];
  atomicAdd(d + 0, v.x);
  atomicAdd(d + 1, v.y);
  atomicAdd(d + 2, v.z);
  atomicAdd(d + 3, v.w);
}

// ---------------------------------------------------------------------------
// Fused GIN MLP: out = post( relu((x+agg)@Wa + ba) @ Wb + bb )
// post = relu then optional BatchNorm (eval-mode affine).
// Wave tile: 16 rows x 64 cols via V_WMMA_F32_16X16X4_F32.
// ---------------------------------------------------------------------------
__global__ __launch_bounds__(256) void gin_mlp_kernel(
    const float* __restrict__ x, const float* __restrict__ agg,
    const float* __restrict__ wa, const float* __restrict__ ba,
    const float* __restrict__ wb, const float* __restrict__ bb,
    const float* __restrict__ bn_g, const float* __restrict__ bn_b,
    const float* __restrict__ bn_m, const float* __restrict__ bn_v,
    int apply_bn, float* __restrict__ out, int n_nodes) {
  __shared__ float wT[DD * TSTR];          // transposed weight [n][k], 17408 B
  __shared__ float tbuf[8 * 16 * TSTR];    // per-wave intermediate tile, 34816 B

  const int tid  = threadIdx.x;
  const int wave = tid >> 5;
  const int lane = tid & 31;
  const int l16  = lane & 15;
  const int hi   = lane >> 4;              // A/B hi-half handles K+2 (ISA 7.12.2)
  const int rowBase = blockIdx.x * 128 + wave * 16;

  // ---- stage Wa^T into LDS ----
  #pragma unroll
  for (int i = 0; i < 16; ++i) {
    int e = tid + i * 256;
    int k = e >> 6, n = e & 63;
    wT[n * TSTR + k] = wa[k * DD + n];
  }
  __syncthreads();

  const v8f vzero = {0.f, 0.f, 0.f, 0.f, 0.f, 0.f, 0.f, 0.f};
  v8f acc[4] = {vzero, vzero, vzero, vzero};

  // ---- GEMM 1: A rows from global (x + agg), B from LDS ----
  int r = rowBase + l16;
  r = (r < n_nodes) ? r : (n_nodes - 1);   // clamp tail (stores predicated later)
  for (int k = 0; k < DD; k += 4) {
    const int kk = k + hi * 2;
    v2f a;
    a.x = x[r * DD + kk]     + agg[r * DD + kk];
    a.y = x[r * DD + kk + 1] + agg[r * DD + kk + 1];
    #pragma unroll
    for (int n = 0; n < 4; ++n) {
      const int col = n * 16 + l16;
      v2f b;
      b.x = wT[col * TSTR + kk];
      b.y = wT[col * TSTR + kk + 1];
      acc[n] = wmma_f32_16x16x4(a, b, acc[n]);
    }
  }

  // bias + ReLU, stage D-layout -> LDS so GEMM2 can refetch in A-layout
  float* tw = &tbuf[wave * 16 * TSTR];
  #pragma unroll
  for (int n = 0; n < 4; ++n) {
    const int col = n * 16 + l16;
    const float bias = ba[col];
    #pragma unroll
    for (int j = 0; j < 8; ++j) {
      float v = acc[n][j] + bias;
      v = v > 0.f ? v : 0.f;
      tw[(j + hi * 8) * TSTR + col] = v;
    }
  }
  __syncthreads();                          // everyone done reading Wa

  // ---- stage Wb^T into LDS (reuse buffer) ----
  #pragma unroll
  for (int i = 0; i < 16; ++i) {
    int e = tid + i * 256;
    int k = e >> 6, n = e & 63;
    wT[n * TSTR + k] = wb[k * DD + n];
  }
  __syncthreads();

  v8f acc2[4] = {vzero, vzero, vzero, vzero};

  // ---- GEMM 2: A from own wave's LDS tile, B from LDS ----
  for (int k = 0; k < DD; k += 4) {
    const int kk = k + hi * 2;
    v2f a;
    a.x = tw[l16 * TSTR + kk];
    a.y = tw[l16 * TSTR + kk + 1];
    #pragma unroll
    for (int n = 0; n < 4; ++n) {
      const int col = n * 16 + l16;
      v2f b;
      b.x = wT[col * TSTR + kk];
      b.y = wT[col * TSTR + kk + 1];
      acc2[n] = wmma_f32_16x16x4(a, b, acc2[n]);
    }
  }

  // ---- bias, ReLU, optional BN, store ----
  #pragma unroll
  for (int n = 0; n < 4; ++n) {
    const int col = n * 16 + l16;
    const float bias = bb[col];
    float scale = 1.f, shift = 0.f;
    if (apply_bn) {
      scale = bn_g[col] * rsqrtf(bn_v[col] + BN_EPS);
      shift = bn_b[col] - bn_m[col] * scale;
    }
    #pragma unroll
    for (int j = 0; j < 8; ++j) {
      const int rr = rowBase + j + hi * 8;
      if (rr < n_nodes) {
        float v = acc2[n][j] + bias;
        v = v > 0.f ? v : 0.f;               // ReLU precedes BN in reference
        out[rr * DD + col] = v * scale + shift;
      }
    }
  }
}

// ---------------------------------------------------------------------------
// Graph pooling: per-graph sum (for mean), count, and bitwise max (h >= 0).
// One thread per (node, float4 chunk).
// ---------------------------------------------------------------------------
__global__ __launch_bounds__(256) void pool_kernel(
    const float4* __restrict__ h4, const int* __restrict__ batch,
    float* __restrict__ meansum, unsigned int* __restrict__ maxbuf,
    float* __restrict__ counts, int n_nodes) {
  int idx = blockIdx.x * 256 + threadIdx.x;     // total = N * 16
  if (idx >= n_nodes * 16) return;
  int i = idx >> 4;
  int c = idx & 15;
  int g = batch[i];
  float4 v = h4[idx];
  float* ms = &meansum[g * DD + c * 4];
  unsigned int* mb = &maxbuf[g * DD + c * 4];
  atomicAdd(ms + 0, v.x);
  atomicAdd(ms + 1, v.y);
  atomicAdd(ms + 2, v.z);
  atomicAdd(ms + 3, v.w);
  atomicMax(mb + 0, __float_as_uint(v.x));      // values >= 0 post-ReLU, so
  atomicMax(mb + 1, __float_as_uint(v.y));      // uint bit-order == float order
  atomicMax(mb + 2, __float_as_uint(v.z));
  atomicMax(mb + 3, __float_as_uint(v.w));
  if (c == 0) atomicAdd(&counts[g], 1.0f);
}

// ---------------------------------------------------------------------------
// Head: pred = ([mean,max] @ Wp1 + bp1) @ Wp2 + bp2 ; log_softmax(2)
// One block (64 threads) per graph.
// ---------------------------------------------------------------------------
__global__ __launch_bounds__(64) void head_kernel(
    const float* __restrict__ meansum, const unsigned int* __restrict__ maxbuf,
    const float* __restrict__ counts,
    const float* __restrict__ wp1, const float* __restrict__ bp1,
    const float* __restrict__ wp2, const float* __restrict__ bp2,
    float* __restrict__ out) {
  __shared__ float gx[2 * DD];
  __shared__ float hid[DD];
  __shared__ float pr[2];
  const int g = blockIdx.x;
  const int t = threadIdx.x;

  float cnt = counts[g];
  cnt = cnt > 1.f ? cnt : 1.f;
  gx[t]      = meansum[g * DD + t] / cnt;
  gx[DD + t] = __uint_as_float(maxbuf[g * DD + t]);  // 0 for empty graphs
  __syncthreads();

  float s = bp1[t];
  #pragma unroll 4
  for (int k = 0; k < 2 * DD; ++k) s += gx[k] * wp1[k * DD + t];
  hid[t] = s;                                        // no activation in head
  __syncthreads();

  if (t < 2) {
    float p = bp2[t];
    for (int k = 0; k < DD; ++k) p += hid[k] * wp2[k * 2 + t];
    pr[t] = p;
  }
  __syncthreads();
  if (t < 2) {
    float m   = pr[0] > pr[1] ? pr[0] : pr[1];
    float lse = m + logf(expf(pr[0] - m) + expf(pr[1] - m));
    out[g * 2 + t] = pr[t] - lse;
  }
}

// ---------------------------------------------------------------------------
extern "C" void kernel_launch(void* const* d_in, const int* in_sizes, int n_in,
                              void* d_out, int out_size, void* d_ws, size_t ws_size,
                              hipStream_t stream) {
  const float* x   = (const float*)d_in[0];
  const int*   ei  = (const int*)d_in[1];
  const int*   bat = (const int*)d_in[2];
  const float* W[6]  = {(const float*)d_in[3],  (const float*)d_in[5],
                        (const float*)d_in[7],  (const float*)d_in[9],
                        (const float*)d_in[11], (const float*)d_in[13]};
  const float* B[6]  = {(const float*)d_in[4],  (const float*)d_in[6],
                        (const float*)d_in[8],  (const float*)d_in[10],
                        (const float*)d_in[12], (const float*)d_in[14]};
  const float* bn0[4] = {(const float*)d_in[15], (const float*)d_in[16],
                         (const float*)d_in[17], (const float*)d_in[18]};
  const float* bn1[4] = {(const float*)d_in[19], (const float*)d_in[20],
                         (const float*)d_in[21], (const float*)d_in[22]};
  const float* wp1 = (const float*)d_in[23];
  const float* bp1 = (const float*)d_in[24];
  const float* wp2 = (const float*)d_in[25];
  const float* bp2 = (const float*)d_in[26];
  float* outp = (float*)d_out;

  const int N = in_sizes[0] / DD;
  const int E = in_sizes[1] / 2;
  const int G = out_size / 2;

  float* f = (float*)d_ws;
  float* bufA    = f;                              // N*DD
  float* bufB    = f + (size_t)N * DD;             // N*DD
  float* agg     = f + (size_t)2 * N * DD;         // N*DD
  float* meansum = f + (size_t)3 * N * DD;         // G*DD
  unsigned int* maxbuf = (unsigned int*)(meansum + (size_t)G * DD);  // G*DD
  float* counts  = (float*)(maxbuf + (size_t)G * DD);                // G

  const int aggBlocks  = (E * 16 + 255) / 256;
  const int mlpBlocks  = (N + 127) / 128;
  const int poolBlocks = (N * 16 + 255) / 256;

  const float* cur = x;
  float* nxt;
  for (int l = 0; l < 3; ++l) {
    hipMemsetAsync(agg, 0, (size_t)N * DD * sizeof(float), stream);
    agg_kernel<<<aggBlocks, 256, 0, stream>>>((const float4*)cur, ei, agg, E);
    nxt = (l == 0) ? bufA : (l == 1) ? bufB : bufA;
    const float** bn = (l == 0) ? bn0 : bn1;       // unused when apply_bn==0
    gin_mlp_kernel<<<mlpBlocks, 256, 0, stream>>>(
        cur, agg, W[2 * l], B[2 * l], W[2 * l + 1], B[2 * l + 1],
        bn[0], bn[1], bn[2], bn[3], (l < 2) ? 1 : 0, nxt, N);
    cur = nxt;
  }

  // pool buffers: meansum | maxbuf | counts are contiguous -> one memset
  hipMemsetAsync(meansum, 0, ((size_t)2 * G * DD + G) * sizeof(float), stream);
  pool_kernel<<<poolBlocks, 256, 0, stream>>>((const float4*)cur, bat, meansum,
                                              maxbuf, counts, N);
  head_kernel<<<G, 64, 0, stream>>>(meansum, maxbuf, counts, wp1, bp1, wp2, bp2, outp);
}